// RandomActivation_24567212933826
// MI455X (gfx1250) — compile-verified
//
#include <hip/hip_runtime.h>

// RandomActivation: out[b, c] = min(act[func_id[c]](x[b, c]), max_output[c])
// x: [16384, 4096] f32, max_output: [4096] f32, func_id: [4096] i32.
// Memory-bound: 512 MiB streamed -> ~23 us floor at 23.3 TB/s.
// This version:
//  - 4 rows x vec4 per thread (64 B in / 64 B out), row stride 16 KiB fits the
//    24-bit immediate offset -> 1 base + 4 clause'd global_load_b128 NT.
//  - per-column func_id compares done ONCE per thread (lane masks in SGPRs),
//    reused by 4 rows of v_cndmask -> fully branchless, no exec-mask branching.
//  - asm "+v" pins stop the compiler from sinking activations into branches.
//  - CDNA5 hardware transcendentals: v_exp_f32, v_rcp_f32, v_tanh_f32.

typedef float v4f __attribute__((ext_vector_type(4)));
typedef int   v4i __attribute__((ext_vector_type(4)));

#ifndef RA_BATCH
#define RA_BATCH 16384
#endif
#ifndef RA_DIM
#define RA_DIM 4096
#endif

#define RA_C4      (RA_DIM / 4)   // 1024 vec4 column-groups per row
#define RA_ROWS_PT 4              // rows per thread
#define RA_LOG2E   1.44269504088896340736f

// Keep a value in a VGPR: forbids the compiler from sinking its computation
// into a divergent branch (forces cndmask selection instead of s_cbranch).
#define RA_PIN(v) asm("" : "+v"(v))

__device__ __forceinline__ float ra_exp(float x) {
  return __builtin_amdgcn_exp2f(x * RA_LOG2E);   // v_exp_f32 is exp2
}

__device__ __forceinline__ float ra_sigmoid(float x) {
  return __builtin_amdgcn_rcpf(1.0f + ra_exp(-x));  // v_exp_f32 + v_rcp_f32
}

__device__ __forceinline__ float ra_tanh(float x) {
#if __has_builtin(__builtin_amdgcn_tanhf)
  return __builtin_amdgcn_tanhf(x);              // gfx1250 v_tanh_f32
#else
  return __builtin_fmaf(2.0f, ra_sigmoid(2.0f * x), -1.0f);
#endif
}

__global__ void __launch_bounds__(256)
RandomActivation_24567212933826_kernel(const float* __restrict__ x,
                                       const float* __restrict__ max_output,
                                       const int*   __restrict__ func_id,
                                       float*       __restrict__ out) {
  // Thread t: column-group c4 = t & 1023, row block r = t >> 10 (4 rows each).
  const unsigned t    = blockIdx.x * 256u + threadIdx.x;
  const unsigned c4   = t & (RA_C4 - 1u);
  const unsigned base = (t & ~(RA_C4 - 1u)) * 4u + c4;   // vec4 index of row 4r

  // Hot 32 KB param tables: cached 128-bit loads.
  const v4f mv = *((const v4f*)max_output + c4);
  const v4i fv = *((const v4i*)func_id + c4);

  // Streaming input: 4 NT 128-bit loads, immediate offsets 0/16/32/48 KiB.
  const v4f* xp = (const v4f*)x + base;
  v4f xs[RA_ROWS_PT];
#pragma unroll
  for (int rr = 0; rr < RA_ROWS_PT; ++rr)
    xs[rr] = __builtin_nontemporal_load(xp + rr * RA_C4);

  v4f rs[RA_ROWS_PT];
#pragma unroll
  for (int k = 0; k < 4; ++k) {          // vec4 component = column
    const int   fid = fv[k];
    const float mo  = mv[k];
    // Per-column masks, computed once, reused across the 4 rows.
    const float sl = (fid == 2) ? 0.01f : 0.0f;  // leaky slope; 0 -> plain relu
    const bool  b1 = (fid == 1);                 // silu
    const bool  b3 = (fid == 3);                 // sigmoid
    const bool  b4 = (fid >= 4);                 // tanh
#pragma unroll
    for (int rr = 0; rr < RA_ROWS_PT; ++rr) {
      const float xx = xs[rr][k];
      float sg   = ra_sigmoid(xx);               // also feeds silu
      float silu = xx * sg;
      float th   = ra_tanh(xx);
      float m    = __builtin_fmaxf(xx, sl * xx); // relu (sl=0) or leaky (sl=.01)
      RA_PIN(sg); RA_PIN(silu); RA_PIN(th); RA_PIN(m);
      float r = m;
      r = b1 ? silu : r;
      r = b3 ? sg   : r;
      r = b4 ? th   : r;
      rs[rr][k] = __builtin_fminf(r, mo);
    }
  }

  // Streaming output: 4 NT 128-bit stores.
  v4f* op = (v4f*)out + base;
#pragma unroll
  for (int rr = 0; rr < RA_ROWS_PT; ++rr)
    __builtin_nontemporal_store(rs[rr], op + rr * RA_C4);
}

extern "C" void kernel_launch(void* const* d_in, const int* in_sizes, int n_in,
                              void* d_out, int out_size, void* d_ws, size_t ws_size,
                              hipStream_t stream) {
  (void)in_sizes; (void)n_in; (void)out_size; (void)d_ws; (void)ws_size;

  const float* x          = (const float*)d_in[0];
  const float* max_output = (const float*)d_in[1];
  const int*   func_id    = (const int*)d_in[2];
  float*       out        = (float*)d_out;

  // Threads: (BATCH / 4 rows) * 1024 col-groups = 4,194,304 -> 16384 blocks.
  const unsigned total_threads = (RA_BATCH / RA_ROWS_PT) * RA_C4;
  const unsigned block = 256;                 // 8 wave32s
  const unsigned grid  = total_threads / block;

  RandomActivation_24567212933826_kernel<<<grid, block, 0, stream>>>(
      x, max_output, func_id, out);
}